// GraphConv_87806311399691
// MI455X (gfx1250) — compile-verified
//
#include <hip/hip_runtime.h>
#include <hip/hip_bf16.h>

typedef __attribute__((ext_vector_type(2))) float v2f;
typedef __attribute__((ext_vector_type(8))) float v8f;

#define N_NODES   50000
#define N_FEAT    128
#define N_EDGES   800000
#define ROW_TILES (N_NODES / 16)   // 3125
#define KP_STRIDE 288              // 256 floats + 32 pad -> halves hit disjoint LDS banks

// ---------------------------------------------------------------------------
// Kernel 1: zero the aggregation buffer (agg lives in d_ws)
// ---------------------------------------------------------------------------
__global__ __launch_bounds__(256) void gc_zero(float4* __restrict__ p, int n4) {
    int i = blockIdx.x * blockDim.x + threadIdx.x;
    if (i < n4) p[i] = make_float4(0.f, 0.f, 0.f, 0.f);
}

// ---------------------------------------------------------------------------
// Kernel 2: scatter  agg[dst] += edge_attr * x[src]
// One wave32 per edge; each lane handles a float4 (128 feats / 32 lanes).
// unsafeAtomicAdd -> global_atomic_add_f32 (HW fp32 atomic at L2).
// ---------------------------------------------------------------------------
__global__ __launch_bounds__(256) void gc_scatter(const float* __restrict__ x,
                                                  const long long* __restrict__ ei,
                                                  const float* __restrict__ ea,
                                                  float* __restrict__ agg, int E) {
    int wave = (blockIdx.x * blockDim.x + threadIdx.x) >> 5;
    int lane = threadIdx.x & 31;
    if (wave >= E) return;
    long long s = ei[wave];
    long long d = ei[(long long)E + wave];
    float e = ea[wave];
    float4 v = *((const float4*)(x + s * N_FEAT) + lane);
    float* a = agg + d * N_FEAT + lane * 4;
    unsafeAtomicAdd(a + 0, e * v.x);
    unsafeAtomicAdd(a + 1, e * v.y);
    unsafeAtomicAdd(a + 2, e * v.z);
    unsafeAtomicAdd(a + 3, e * v.w);
}

// ---------------------------------------------------------------------------
// Kernel 3: out = agg @ W_rel^T + x @ W_root^T + b  as one K=256 WMMA GEMM.
// B[k][n] = W_rel[n][k] (k<128) | W_root[n][k-128], staged in LDS as
// Bs[kpair*288 + n*2 + (k&1)]  -> ds_load_b64 B-fragments, bank-conflict-free.
// Wave layout (V_WMMA_F32_16X16X4_F32):
//   A frag: lane l holds A[row=base+l%16][k0+2h], A[..][k0+2h+1], h=(l>=16)
//   B frag: lane l holds B[k0+2h][n=l%16],       B[k0+2h+1][n]
//   C/D   : vgpr r -> M = r + 8h, N = l%16
// ---------------------------------------------------------------------------
__global__ __launch_bounds__(128) void gc_gemm(const float* __restrict__ x,
                                               const float* __restrict__ agg,
                                               const float* __restrict__ Wrel,
                                               const float* __restrict__ Wroot,
                                               const float* __restrict__ bias,
                                               float* __restrict__ out, int nRowTiles) {
    __shared__ float Bs[128 * KP_STRIDE];   // 128 kpairs * 288 floats = 147456 B

    const int tid = threadIdx.x;
    // Stage combined weights into LDS (coalesced global reads).
    for (int idx = tid; idx < 256 * 128; idx += 128) {
        int n = idx >> 8;          // 0..127
        int k = idx & 255;         // 0..255
        float w = (k < 128) ? Wrel[n * 128 + k] : Wroot[n * 128 + (k - 128)];
        Bs[(k >> 1) * KP_STRIDE + n * 2 + (k & 1)] = w;
    }
    __syncthreads();

    const int waveId  = tid >> 5;
    const int lane15  = tid & 15;
    const int h       = (tid >> 4) & 1;          // half-wave select
    const int rowTile = blockIdx.x * 4 + waveId; // uniform per wave

    if (rowTile < nRowTiles) {
        const int rowBase = rowTile * 16;
        const int row     = rowBase + lane15;
        const float* arow = agg + (size_t)row * N_FEAT;
        const float* xrow = x   + (size_t)row * N_FEAT;

        v8f acc[8];
        #pragma unroll
        for (int t = 0; t < 8; ++t) acc[t] = (v8f)(0.f);

        // K = 0..127 : A = agg, B = W_rel^T
        for (int kt = 0; kt < 32; ++kt) {
            v2f a = *(const v2f*)(arow + kt * 4 + 2 * h);
            const float* bp = &Bs[(kt * 2 + h) * KP_STRIDE];
            #pragma unroll
            for (int t = 0; t < 8; ++t) {
                v2f bb = *(const v2f*)(bp + (t * 16 + lane15) * 2);
                acc[t] = __builtin_amdgcn_wmma_f32_16x16x4_f32(
                    false, a, false, bb, (short)0, acc[t], false, false);
            }
        }
        // K = 128..255 : A = x, B = W_root^T
        for (int kt = 0; kt < 32; ++kt) {
            v2f a = *(const v2f*)(xrow + kt * 4 + 2 * h);
            const float* bp = &Bs[((kt + 32) * 2 + h) * KP_STRIDE];
            #pragma unroll
            for (int t = 0; t < 8; ++t) {
                v2f bb = *(const v2f*)(bp + (t * 16 + lane15) * 2);
                acc[t] = __builtin_amdgcn_wmma_f32_16x16x4_f32(
                    false, a, false, bb, (short)0, acc[t], false, false);
            }
        }

        // Bias add + store. vgpr r -> row rowBase + r + 8h, col n.
        #pragma unroll
        for (int t = 0; t < 8; ++t) {
            int n = t * 16 + lane15;
            float bv = bias[n];
            #pragma unroll
            for (int r = 0; r < 8; ++r) {
                out[(size_t)(rowBase + r + 8 * h) * N_FEAT + n] = acc[t][r] + bv;
            }
        }
    }
}

// ---------------------------------------------------------------------------
extern "C" void kernel_launch(void* const* d_in, const int* in_sizes, int n_in,
                              void* d_out, int out_size, void* d_ws, size_t ws_size,
                              hipStream_t stream) {
    const float*     x     = (const float*)d_in[0];
    const long long* ei    = (const long long*)d_in[1];
    const float*     ea    = (const float*)d_in[2];
    const float*     Wrel  = (const float*)d_in[3];
    const float*     Wroot = (const float*)d_in[4];
    const float*     bias  = (const float*)d_in[5];
    float*           out   = (float*)d_out;
    float*           agg   = (float*)d_ws;   // N_NODES*N_FEAT floats = 25.6 MB

    const int E = in_sizes[2];                      // 800000
    const int n4 = N_NODES * N_FEAT / 4;            // 1.6M float4

    gc_zero<<<(n4 + 255) / 256, 256, 0, stream>>>((float4*)agg, n4);
    gc_scatter<<<(E * 32 + 255) / 256, 256, 0, stream>>>(x, ei, ea, agg, E);
    gc_gemm<<<(ROW_TILES + 3) / 4, 128, 0, stream>>>(x, agg, Wrel, Wroot, bias, out,
                                                     ROW_TILES);
}